// MultiUMPS_82764019793998
// MI455X (gfx1250) — compile-verified
//
#include <hip/hip_runtime.h>
#include <math.h>

typedef __attribute__((ext_vector_type(2))) float v2f;
typedef __attribute__((ext_vector_type(8))) float v8f;

#define BOND    64
#define FEAT    31
#define OUT_DIM 16
#define N_UMPS  4
#define B_SZ    16
#define LSEQ    1024
#define HALF_L  512
#define TCH     4            // timesteps formed per chunk
#define NS_ITERS 4           // Newton-Schulz iterations for tr(sqrt(M^T M))
#define MAT     4096         // 64*64

// ---- dynamic LDS layout (in floats) ----
#define OFF_M    0                         // M chunk: TCH * 4096
#define OFF_P    (OFF_M + TCH * MAT)       // running product
#define OFF_P2   (OFF_P + MAT)             // product double-buffer
#define OFF_A    (OFF_P2 + MAT)            // NS buffer A (G / Y)
#define OFF_B    (OFF_A + MAT)             // NS buffer B (Z)
#define OFF_C    (OFF_B + MAT)             // NS buffer C (T/W)
#define OFF_D    (OFF_C + MAT)             // NS buffer D (U)
#define OFF_X    (OFF_D + MAT)             // x chunk: TCH * 32
#define OFF_RED  (OFF_X + TCH * 32)        // reduction scratch: 256
#define OFF_V    (OFF_RED + 256)           // alpha^T P vector: 64
#define SMEM_FLOATS (OFF_V + 64)           // = 41408 floats = 165,632 bytes

// Workgroup-wide 64x64x64 matmul: C = A @ B (or A^T @ B), f32 WMMA 16x16x4.
// 256 threads = 8 wave32; 16 output tiles of 16x16 -> 2 tiles per wave.
// Fragment layouts per CDNA5 ISA (CDNA5 ISA):
//   A 16x4 f32:  lanes 0-15 hold K={k0,k0+1}, lanes 16-31 hold K={k0+2,k0+3}
//   B 4x16 f32:  same K split; N striped across the 16 lanes of each half
//   C/D 16x16:   VGPR v: lanes 0-15 -> M=v, lanes 16-31 -> M=v+8
template <bool TRANSA>
__device__ __forceinline__ void wg_mm64(float* __restrict__ C,
                                        const float* __restrict__ A,
                                        const float* __restrict__ B) {
  const int tid  = threadIdx.x;
  const int lane = tid & 31;
  const int wave = tid >> 5;
  const int half = lane >> 4;   // 0: lanes 0-15, 1: lanes 16-31
  const int l16  = lane & 15;
  __syncthreads();              // all producers of A/B done
#pragma unroll
  for (int rep = 0; rep < 2; ++rep) {
    const int tt = wave * 2 + rep;          // tile id 0..15
    const int tm = (tt >> 2) << 4;          // tile row base
    const int tn = (tt & 3) << 4;           // tile col base
    v8f acc = {};
#pragma unroll
    for (int kb = 0; kb < 16; ++kb) {
      const int k0 = (kb << 2) + (half << 1);
      const int m  = tm + l16;
      v2f a, b;
      if (TRANSA) {
        a[0] = A[(k0)     * 64 + m];
        a[1] = A[(k0 + 1) * 64 + m];
      } else {
        a[0] = A[m * 64 + k0];
        a[1] = A[m * 64 + k0 + 1];
      }
      b[0] = B[(k0)     * 64 + tn + l16];
      b[1] = B[(k0 + 1) * 64 + tn + l16];
      acc = __builtin_amdgcn_wmma_f32_16x16x4_f32(false, a, false, b,
                                                  (short)0, acc, false, false);
    }
    const int r0 = tm + (half << 3);
#pragma unroll
    for (int v = 0; v < 8; ++v)
      C[(r0 + v) * 64 + tn + l16] = acc[v];
  }
  __syncthreads();              // consumers may read C
}

__device__ __forceinline__ float block_reduce_sum(float v, float* red, int tid) {
  __syncthreads();
  red[tid] = v;
  __syncthreads();
#pragma unroll
  for (int s = 128; s > 0; s >>= 1) {
    if (tid < s) red[tid] += red[tid + s];
    __syncthreads();
  }
  const float r = red[0];
  __syncthreads();
  return r;
}

// One workgroup per (batch, ump) chain: 64 workgroups total.
__global__ void __launch_bounds__(256)
umps_chain_kernel(const float* __restrict__ inputs,   // (16,1024,32)
                  const float* __restrict__ tcores,   // (4,64,31,64)
                  const float* __restrict__ alphas,   // (4,64)
                  const float* __restrict__ omegas,   // (4,64)
                  const float* __restrict__ ocores,   // (4,64,16,64)
                  float* __restrict__ feats)          // (16,64)
{
  extern __shared__ float sm[];
  const int tid   = threadIdx.x;
  const int chain = blockIdx.x;
  const int bb    = chain >> 2;   // batch
  const int u     = chain & 3;    // ump index

  float* Mbuf = sm + OFF_M;
  float* P    = sm + OFF_P;
  float* P2   = sm + OFF_P2;
  float* bufA = sm + OFF_A;
  float* bufB = sm + OFF_B;
  float* bufC = sm + OFF_C;
  float* bufD = sm + OFF_D;
  float* xs   = sm + OFF_X;
  float* red  = sm + OFF_RED;
  float* vbuf = sm + OFF_V;

  // running product P = I
  for (int e = tid; e < MAT; e += 256)
    P[e] = ((e >> 6) == (e & 63)) ? 1.0f : 0.0f;

  float* Pc = P;
  float* Pn = P2;

  for (int t0 = 0; t0 < HALF_L; t0 += TCH) {
    // ---- load x chunk ----
    __syncthreads();
    if (tid < TCH * 32) {
      const int t = tid >> 5, f = tid & 31;
      xs[tid] = inputs[((size_t)(bb * LSEQ + t0 + t)) * 32 + f];
    }
    __syncthreads();

    // ---- form M chunk: M[t][i][j] = x[t,0]*delta_ij + sum_f x[t,f]*tc[i,f-1,j] ----
    for (int e = 0; e < 16; ++e) {
      const int pp = e * 256 + tid;          // consecutive tid -> consecutive j
      const int i = pp >> 6, j = pp & 63;
      const float diag = (i == j) ? 1.0f : 0.0f;
      float a0 = xs[0]      * diag;
      float a1 = xs[32]     * diag;
      float a2 = xs[64]     * diag;
      float a3 = xs[96]     * diag;
      const float* tc = tcores + ((size_t)(u * 64 + i)) * FEAT * 64 + j;
      for (int f = 1; f < 32; ++f) {
        const float cv = tc[(size_t)(f - 1) * 64];
        a0 += xs[f]      * cv;
        a1 += xs[32 + f] * cv;
        a2 += xs[64 + f] * cv;
        a3 += xs[96 + f] * cv;
      }
      Mbuf[0 * MAT + pp] = a0;
      Mbuf[1 * MAT + pp] = a1;
      Mbuf[2 * MAT + pp] = a2;
      Mbuf[3 * MAT + pp] = a3;
    }

    // ---- per timestep: nuclear-norm normalize, then chain multiply ----
    for (int ts = 0; ts < TCH; ++ts) {
      float* Mt = Mbuf + ts * MAT;

      // G = Mt^T @ Mt
      wg_mm64<true>(bufA, Mt, Mt);

      // s = ||G||_F  (>= spectral norm -> valid Newton-Schulz scaling)
      float ssq = 0.0f;
      for (int e = 0; e < 16; ++e) {
        const float g = bufA[e * 256 + tid];
        ssq += g * g;
      }
      const float s = sqrtf(block_reduce_sum(ssq, red, tid));
      const float inv_s = (s > 1e-30f) ? (1.0f / s) : 0.0f;

      // Y0 = G/s (in place), Z0 = I
      for (int e = tid; e < MAT; e += 256) {
        bufA[e] *= inv_s;
        bufB[e] = ((e >> 6) == (e & 63)) ? 1.0f : 0.0f;
      }

      float *pY = bufA, *pZ = bufB, *pT = bufC, *pU = bufD;
      for (int it = 0; it < NS_ITERS; ++it) {
        wg_mm64<false>(pT, pZ, pY);          // T = Z @ Y
        for (int e = tid; e < MAT; e += 256) {  // T <- 0.5*(3I - T)
          float w = -0.5f * pT[e];
          if ((e >> 6) == (e & 63)) w += 1.5f;
          pT[e] = w;
        }
        wg_mm64<false>(pU, pY, pT);          // next Y
        wg_mm64<false>(pY, pT, pZ);          // next Z (into old-Y buffer)
        float* oy = pY; float* oz = pZ;
        pY = pU; pZ = oy; pU = oz;           // rotate (pT reused as-is)
      }

      // nuclear(M) = sqrt(s) * tr(Y);  nuc = nuclear/64;  Mt *= 1/nuc
      const float dv  = (tid < 64) ? pY[tid * 64 + tid] : 0.0f;
      const float trY = block_reduce_sum(dv, red, tid);
      const float den = sqrtf(s) * trY;
      const float fac = (fabsf(den) > 1e-30f) ? (64.0f / den) : 1.0f;
      for (int e = tid; e < MAT; e += 256) Mt[e] *= fac;

      // Pn = Pc @ Mt
      wg_mm64<false>(Pn, Pc, Mt);
      float* tp = Pc; Pc = Pn; Pn = tp;
    }
  }

  // ---- epilogue: out[l] = sum_k (alpha^T P)[k] * sum_m oc[k,l,m]*omega[m] ----
  __syncthreads();
  if (tid < 64) {
    float v = 0.0f;
    for (int j = 0; j < 64; ++j) v += alphas[u * 64 + j] * Pc[j * 64 + tid];
    vbuf[tid] = v;
  }
  __syncthreads();
  {
    const int l = tid & 15, kg = tid >> 4;   // 16 k-groups of 4
    float part = 0.0f;
    for (int kk = 0; kk < 4; ++kk) {
      const int k = kg * 4 + kk;
      const float* oc = ocores + ((size_t)((u * 64 + k) * 16 + l)) * 64;
      float inner = 0.0f;
      for (int m = 0; m < 64; ++m) inner += oc[m] * omegas[u * 64 + m];
      part += vbuf[k] * inner;
    }
    red[tid] = part;
  }
  __syncthreads();
  if (tid < 16) {
    float acc = 0.0f;
    for (int g = 0; g < 16; ++g) acc += red[g * 16 + tid];
    feats[bb * 64 + u * 16 + tid] = acc;     // concat order: ump-major per batch
  }
}

// feats (16,64) @ W_out (64,16) + b_out -> (16,16)
__global__ void __launch_bounds__(256)
final_linear_kernel(const float* __restrict__ feats,
                    const float* __restrict__ W_out,
                    const float* __restrict__ b_out,
                    float* __restrict__ out)
{
  const int tid = threadIdx.x;
  const int o = tid & 15, b = tid >> 4;
  float acc = b_out[o];
  for (int c = 0; c < 64; ++c) acc += feats[b * 64 + c] * W_out[c * 16 + o];
  out[b * 16 + o] = acc;
}

extern "C" void kernel_launch(void* const* d_in, const int* in_sizes, int n_in,
                              void* d_out, int out_size, void* d_ws, size_t ws_size,
                              hipStream_t stream) {
  (void)in_sizes; (void)n_in; (void)out_size; (void)ws_size;
  const float* inputs = (const float*)d_in[0];
  const float* tcores = (const float*)d_in[1];
  const float* alphas = (const float*)d_in[2];
  const float* omegas = (const float*)d_in[3];
  const float* ocores = (const float*)d_in[4];
  const float* W_out  = (const float*)d_in[5];
  const float* b_out  = (const float*)d_in[6];
  float* feats = (float*)d_ws;               // 16*64 floats of scratch
  float* out   = (float*)d_out;

  const size_t smem = (size_t)SMEM_FLOATS * sizeof(float);  // ~162 KB (<320 KB/WGP)
  (void)hipFuncSetAttribute((const void*)umps_chain_kernel,
                            hipFuncAttributeMaxDynamicSharedMemorySize, (int)smem);

  umps_chain_kernel<<<dim3(B_SZ * N_UMPS), dim3(256), smem, stream>>>(
      inputs, tcores, alphas, omegas, ocores, feats);
  final_linear_kernel<<<dim3(1), dim3(256), 0, stream>>>(feats, W_out, b_out, out);
}